// ReadUnit_41008347742335
// MI455X (gfx1250) — compile-verified
//
#include <hip/hip_runtime.h>
#include <hip/hip_bf16.h>

typedef __attribute__((ext_vector_type(2))) float v2f;
typedef __attribute__((ext_vector_type(8))) float v8f;

#define T_  2
#define B_  256
#define DIM_ 512
#define N_  196

// ---------------------------------------------------------------------------
// Wave-per-strip f32 WMMA GEMM:  C[M,N] = (A .* scale?) @ B (+ bias?)
//   A  : [M,K] row-major
//   Bm : B_TRANS ? [N,K] row-major (access B(k,n)=Bm[n*K+k])
//                : [K,N] row-major
//   One wave (32 threads) computes a 64x16 output strip, K stepped by 4 using
//   V_WMMA_F32_16X16X4_F32 (full f32 precision, no downconvert).
//   K/N are compile-time so the k-loop has a static trip count (no peel path).
// ---------------------------------------------------------------------------
template <int K, int N, bool B_TRANS, bool A_SCALE, bool HAS_BIAS>
__global__ __launch_bounds__(32)
void wmma_gemm_f32(const float* __restrict__ A,
                   const float* __restrict__ Bm,
                   const float* __restrict__ scale,   // len K (or nullptr)
                   const float* __restrict__ bias,    // len N (or nullptr)
                   float* __restrict__ C) {
    const int lane = threadIdx.x & 31;
    const int half = lane >> 4;          // 0: lanes 0-15, 1: lanes 16-31
    const int lm   = lane & 15;
    constexpr int ntiles = N >> 4;
    const int n0 = (blockIdx.x % ntiles) << 4;
    const int m0 = (blockIdx.x / ntiles) << 6;   // 4 M-tiles per wave

    v8f acc0 = {}; v8f acc1 = {}; v8f acc2 = {}; v8f acc3 = {};

    for (int k0 = 0; k0 < K; k0 += 4) {
        const int ka = k0 + 2 * half;    // this lane's two consecutive k

        // B fragment: b[j] = B(k0 + j + 2*half, n0+lm)
        v2f b;
        if (B_TRANS) {
            b = *(const v2f*)(Bm + (size_t)(n0 + lm) * K + ka);
        } else {
            b.x = Bm[(size_t)(ka + 0) * N + n0 + lm];
            b.y = Bm[(size_t)(ka + 1) * N + n0 + lm];
        }

        // A fragments: a[j] = A[m0 + 16*i + lm][ka + j]  (8B-aligned pair)
        v2f a0 = *(const v2f*)(A + (size_t)(m0 +  0 + lm) * K + ka);
        v2f a1 = *(const v2f*)(A + (size_t)(m0 + 16 + lm) * K + ka);
        v2f a2 = *(const v2f*)(A + (size_t)(m0 + 32 + lm) * K + ka);
        v2f a3 = *(const v2f*)(A + (size_t)(m0 + 48 + lm) * K + ka);
        if (A_SCALE) {
            v2f s = *(const v2f*)(scale + ka);
            a0 *= s; a1 *= s; a2 *= s; a3 *= s;
        }

        acc0 = __builtin_amdgcn_wmma_f32_16x16x4_f32(false, a0, false, b, (short)0, acc0, false, false);
        acc1 = __builtin_amdgcn_wmma_f32_16x16x4_f32(false, a1, false, b, (short)0, acc1, false, false);
        acc2 = __builtin_amdgcn_wmma_f32_16x16x4_f32(false, a2, false, b, (short)0, acc2, false, false);
        acc3 = __builtin_amdgcn_wmma_f32_16x16x4_f32(false, a3, false, b, (short)0, acc3, false, false);
    }

    const int   col = n0 + lm;
    const float bv  = HAS_BIAS ? bias[col] : 0.0f;
#pragma unroll
    for (int r = 0; r < 8; ++r) {
        const int row = r + 8 * half;    // D layout: VGPR r <-> rows r, r+8
        C[(size_t)(m0 +  0 + row) * N + col] = acc0[r] + bv;
        C[(size_t)(m0 + 16 + row) * N + col] = acc1[r] + bv;
        C[(size_t)(m0 + 32 + row) * N + col] = acc2[r] + bv;
        C[(size_t)(m0 + 48 + row) * N + col] = acc3[r] + bv;
    }
}

// ---------------------------------------------------------------------------
// Fused streaming pass: one block per batch row b.
//   w[k]       = mem[b,k]*u[b,k] + u[b,k+512]        (built in LDS)
//   logits[n]  = sum_k w[k] * know[b,k,n]
//   colsum[n]  = sum_k know[b,k,n]
//   out[b,n]   = softmax_n(logits)[n] * colsum[n]
// know is read exactly once (102.8 MB total) -> HBM-bound (~4.4 us floor).
// ---------------------------------------------------------------------------
__global__ __launch_bounds__(256)
void fused_attend(const float* __restrict__ know,   // [B, DIM, N]
                  const float* __restrict__ memv,   // [B, DIM]
                  const float* __restrict__ u,      // [B, 2*DIM]
                  float* __restrict__ out) {        // [B, N]
    const int b = blockIdx.x;
    const int t = threadIdx.x;

    __shared__ float w[DIM_];
    __shared__ float red[256];

    // Build gate vector w[b,:] in LDS.
    for (int k = t; k < DIM_; k += 256)
        w[k] = memv[(size_t)b * DIM_ + k] * u[(size_t)b * (2 * DIM_) + k]
             + u[(size_t)b * (2 * DIM_) + DIM_ + k];
    __syncthreads();

    float lg = 0.0f, cs = 0.0f;
    if (t < N_) {
        const float* kp = know + (size_t)b * DIM_ * N_ + t;
        for (int kc = 0; kc < DIM_; kc += 32) {
            if (kc + 96 < DIM_)   // speculative prefetch ahead (global_prefetch_b8)
                __builtin_prefetch(kp + (size_t)(kc + 96) * N_, 0, 0);
#pragma unroll
            for (int k = kc; k < kc + 32; ++k) {
                const float v = kp[(size_t)k * N_];
                lg = fmaf(w[k], v, lg);
                cs += v;
            }
        }
    }

    // Block max over active lanes.
    red[t] = (t < N_) ? lg : -__builtin_inff();
    __syncthreads();
    for (int s = 128; s > 0; s >>= 1) {
        if (t < s) red[t] = fmaxf(red[t], red[t + s]);
        __syncthreads();
    }
    const float mx = red[0];
    __syncthreads();

    // Block sum of exp.
    const float e = (t < N_) ? expf(lg - mx) : 0.0f;
    red[t] = e;
    __syncthreads();
    for (int s = 128; s > 0; s >>= 1) {
        if (t < s) red[t] += red[t + s];
        __syncthreads();
    }
    const float denom = red[0];

    if (t < N_)
        out[(size_t)b * N_ + t] = (e / denom) * cs;
}

extern "C" void kernel_launch(void* const* d_in, const int* in_sizes, int n_in,
                              void* d_out, int out_size, void* d_ws, size_t ws_size,
                              hipStream_t stream) {
    (void)in_sizes; (void)n_in; (void)out_size; (void)ws_size;

    const float* memory   = (const float*)d_in[0];   // [T,B,DIM]
    const float* know     = (const float*)d_in[1];   // [B,DIM,N]
    const float* control  = (const float*)d_in[2];   // [T,B,DIM]
    const float* W_mem    = (const float*)d_in[3];   // [DIM,DIM]
    const float* b_mem    = (const float*)d_in[4];   // [DIM]
    const float* W_concat = (const float*)d_in[5];   // [2*DIM,DIM]
    // d_in[6] = b_concat, d_in[7] = W_attn, d_in[8] = b_attn
    const float* W_attn   = (const float*)d_in[7];   // [DIM,1]
    // b_concat / b_attn contribute a per-row constant to logits only ->
    // exactly cancelled by softmax shift-invariance; omitted.

    float* mem_ws = (float*)d_ws;                    // [B, DIM]    (512 KB)
    float* u_ws   = mem_ws + (size_t)B_ * DIM_;      // [B, 2*DIM]  (1 MB)

    const float* A_mem  = memory  + (size_t)(T_ - 1) * B_ * DIM_;  // memory[-1]
    const float* A_ctrl = control + (size_t)(T_ - 1) * B_ * DIM_;  // control[-1]

    // mem = memory[-1] @ W_mem + b_mem        (B-matrix row-major [K,N])
    wmma_gemm_f32<DIM_, DIM_, false, false, true>
        <<<(B_ / 64) * (DIM_ / 16), 32, 0, stream>>>(
            A_mem, W_mem, nullptr, b_mem, mem_ws);

    // u = (control[-1] .* W_attn) @ W_concat^T (B stored [N,K] row-major)
    wmma_gemm_f32<DIM_, 2 * DIM_, true, true, false>
        <<<(B_ / 64) * ((2 * DIM_) / 16), 32, 0, stream>>>(
            A_ctrl, W_concat, W_attn, nullptr, u_ws);

    // Single pass over `know`: logits + colsum + softmax + output.
    fused_attend<<<B_, 256, 0, stream>>>(know, mem_ws, u_ws, (float*)d_out);
}